// GNNpool_65274912965021
// MI455X (gfx1250) — compile-verified
//
#include <hip/hip_runtime.h>
#include <math.h>

typedef __attribute__((ext_vector_type(2))) float v2f;
typedef __attribute__((ext_vector_type(8))) float v8f;

__device__ __forceinline__ float elu1(float x) { return x > 0.f ? x : expm1f(x); }

// ---------------- degree / norm ----------------
__global__ void k_deg_init(float* deg, int n) {
    int i = blockIdx.x * blockDim.x + threadIdx.x;
    if (i < n) deg[i] = 1.0f;               // self-loop contributes 1
}
__global__ void k_deg_edges(const int* __restrict__ dst, float* deg, int e) {
    int i = blockIdx.x * blockDim.x + threadIdx.x;
    if (i < e) atomicAdd(&deg[dst[i]], 1.0f);
}
__global__ void k_dinv(const float* __restrict__ deg, float* dinv, int n) {
    int i = blockIdx.x * blockDim.x + threadIdx.x;
    if (i < n) dinv[i] = rsqrtf(deg[i]);    // deg >= 1 always
}

// ---------------- layer-1 GEMM (K=3, scalar) : hs = dinv[row] * (x @ W1) ----------------
__global__ void k_gemm_l1(const float* __restrict__ x, const float* __restrict__ W1,
                          const float* __restrict__ dinv,
                          float* __restrict__ hs, float* __restrict__ acc, int n) {
    int i = blockIdx.x * blockDim.x + threadIdx.x;
    if (i >= n * 64) return;
    int node = i >> 6, c = i & 63;
    float s = x[node*3+0]*W1[c] + x[node*3+1]*W1[64+c] + x[node*3+2]*W1[128+c];
    s *= dinv[node];
    hs[i]  = s;   // pre-scaled messages
    acc[i] = s;   // accumulator seeded with self-loop term
}

// ---------------- WMMA f32 GEMM: [n,64] @ [64,NCOLS] ----------------
// MODE 0: GCN layer -> out0=hs=dinv*D, out1=acc=same
// MODE 1: MLP1      -> out0 = elu(D + bias)
template<int NCOLS, int MODE>
__global__ __launch_bounds__(256) void k_gemm_wmma(
    const float* __restrict__ H, const float* __restrict__ W,
    const float* __restrict__ dinv, const float* __restrict__ bias,
    float* __restrict__ out0, float* __restrict__ out1, int n)
{
    __shared__ float sW[64 * NCOLS];
    __shared__ float sH[32 * 64];
    int tid = threadIdx.x;
    for (int i = tid; i < 64 * NCOLS; i += 256) sW[i] = W[i];
    int m0 = blockIdx.x * 32;
    for (int i = tid; i < 32 * 64; i += 256) {
        int gm = m0 + (i >> 6);
        sH[i] = (gm < n) ? H[(long long)gm * 64 + (i & 63)] : 0.f;
    }
    __syncthreads();

    int w = tid >> 5, lane = tid & 31;
    int llo = lane & 15, lhi = lane >> 4;         // lhi: half-wave select
    const int NT = 2 * (NCOLS / 16);              // 2 row-tiles x (NCOLS/16) col-tiles
    for (int tile = w; tile < NT; tile += 8) {
        int mi = tile & 1, ni = tile >> 1;
        v8f c = {0.f,0.f,0.f,0.f,0.f,0.f,0.f,0.f};
        #pragma unroll
        for (int k4 = 0; k4 < 16; ++k4) {
            // A 16x4: lanes<16 hold K={0,1}, lanes>=16 hold K={2,3} (v0=even K, v1=odd K)
            int kk = k4 * 4 + lhi * 2;
            v2f a, b;
            a.x = sH[(mi*16 + llo) * 64 + kk];
            a.y = sH[(mi*16 + llo) * 64 + kk + 1];
            // B 4x16: row K striped across lanes within a VGPR, same K split per half-wave
            b.x = sW[(kk    ) * NCOLS + ni*16 + llo];
            b.y = sW[(kk + 1) * NCOLS + ni*16 + llo];
            c = __builtin_amdgcn_wmma_f32_16x16x4_f32(
                    false, a, false, b, (short)0, c, false, false);
        }
        // D 16x16: VGPR r -> M=r (lanes 0-15), M=r+8 (lanes 16-31); N=lane%16
        #pragma unroll
        for (int r = 0; r < 8; ++r) {
            int m  = mi*16 + r + lhi*8;
            int gm = m0 + m;
            if (gm < n) {
                int col = ni*16 + llo;
                float v = c[r];
                if (MODE == 0) {
                    v *= dinv[gm];
                    out0[(long long)gm*64 + col] = v;
                    out1[(long long)gm*64 + col] = v;
                } else {
                    out0[(long long)gm*NCOLS + col] = elu1(v + bias[col]);
                }
            }
        }
    }
}

// ---------------- edge aggregation: acc[dst] += hs[src], 16 threads/edge ----------------
__global__ void k_edge_agg(const int* __restrict__ src, const int* __restrict__ dst,
                           const float* __restrict__ hs, float* __restrict__ acc, int e) {
    long long i = (long long)blockIdx.x * blockDim.x + threadIdx.x;
    long long ee = i >> 4;
    int r = (int)(i & 15);
    if (ee >= e) return;
    int s = src[ee], d = dst[ee];
    const float4 v = *(const float4*)(hs + (long long)s * 64 + r * 4);
    float* o = acc + (long long)d * 64 + r * 4;
    atomicAdd(o + 0, v.x); atomicAdd(o + 1, v.y);
    atomicAdd(o + 2, v.z); atomicAdd(o + 3, v.w);
}

// ---------------- finalize: h = elu(dinv[node]*acc + b) ----------------
__global__ void k_finalize(const float* __restrict__ acc, const float* __restrict__ dinv,
                           const float* __restrict__ b, float* __restrict__ h, int n) {
    int i = blockIdx.x * blockDim.x + threadIdx.x;
    if (i >= n * 64) return;
    int node = i >> 6, c = i & 63;
    h[i] = elu1(dinv[node] * acc[i] + b[c]);
}

// ---------------- MLP head: [n,128]@[128,15] + bias, softmax ----------------
__global__ __launch_bounds__(256) void k_mlp2_softmax(
    const float* __restrict__ t, const float* __restrict__ Wm2,
    const float* __restrict__ bm2, float* __restrict__ out, int n)
{
    __shared__ float sW[128 * 15];
    __shared__ float sb[15];
    for (int i = threadIdx.x; i < 128 * 15; i += 256) sW[i] = Wm2[i];
    if (threadIdx.x < 15) sb[threadIdx.x] = bm2[threadIdx.x];
    __syncthreads();
    int node = blockIdx.x * 256 + threadIdx.x;
    if (node >= n) return;
    float a[15];
    #pragma unroll
    for (int j = 0; j < 15; ++j) a[j] = sb[j];
    const float* tr = t + (long long)node * 128;
    for (int k = 0; k < 128; ++k) {
        float tv = tr[k];
        #pragma unroll
        for (int j = 0; j < 15; ++j) a[j] = fmaf(tv, sW[k*15 + j], a[j]);
    }
    float m = a[0];
    #pragma unroll
    for (int j = 1; j < 15; ++j) m = fmaxf(m, a[j]);
    float sum = 0.f;
    #pragma unroll
    for (int j = 0; j < 15; ++j) { a[j] = expf(a[j] - m); sum += a[j]; }
    float inv = 1.f / sum;
    float* orow = out + (long long)node * 15;
    #pragma unroll
    for (int j = 0; j < 15; ++j) orow[j] = a[j] * inv;
}

extern "C" void kernel_launch(void* const* d_in, const int* in_sizes, int n_in,
                              void* d_out, int out_size, void* d_ws, size_t ws_size,
                              hipStream_t stream) {
    (void)n_in; (void)out_size; (void)ws_size;
    const float* x   = (const float*)d_in[0];
    const int*   ei  = (const int*)  d_in[1];
    const float* W1  = (const float*)d_in[2];
    const float* b1  = (const float*)d_in[3];
    const float* W2  = (const float*)d_in[4];
    const float* b2  = (const float*)d_in[5];
    const float* Wm1 = (const float*)d_in[6];
    const float* bm1 = (const float*)d_in[7];
    const float* Wm2 = (const float*)d_in[8];
    const float* bm2 = (const float*)d_in[9];

    int n = in_sizes[0] / 3;
    int e = in_sizes[1] / 2;
    const int* src = ei;         // edge_index[0,:]
    const int* dst = ei + e;     // edge_index[1,:]

    float* ws   = (float*)d_ws;
    float* deg  = ws;
    float* dinv = ws + (size_t)n;
    float* A    = ws + (size_t)2 * n;        // hs  [n,64]
    float* B    = A  + (size_t)64 * n;       // acc [n,64]
    float* C    = B  + (size_t)64 * n;       // h   [n,64]
    float* T    = A;                         // mlp hidden [n,128], overlays A+B (free by then)

    dim3 blk(256);
    int gn    = (n + 255) / 256;
    int ge    = (e + 255) / 256;
    int gn64  = (n * 64 + 255) / 256;
    int gedge = (int)(((long long)e * 16 + 255) / 256);
    int gm32  = (n + 31) / 32;

    k_deg_init <<<gn, blk, 0, stream>>>(deg, n);
    k_deg_edges<<<ge, blk, 0, stream>>>(dst, deg, e);
    k_dinv     <<<gn, blk, 0, stream>>>(deg, dinv, n);

    // layer 1
    k_gemm_l1  <<<gn64,  blk, 0, stream>>>(x, W1, dinv, A, B, n);
    k_edge_agg <<<gedge, blk, 0, stream>>>(src, dst, A, B, e);
    k_finalize <<<gn64,  blk, 0, stream>>>(B, dinv, b1, C, n);

    // layer 2 (WMMA f32 GEMM)
    k_gemm_wmma<64, 0><<<gm32, blk, 0, stream>>>(C, W2, dinv, nullptr, A, B, n);
    k_edge_agg <<<gedge, blk, 0, stream>>>(src, dst, A, B, e);
    k_finalize <<<gn64,  blk, 0, stream>>>(B, dinv, b2, C, n);

    // MLP head
    k_gemm_wmma<128, 1><<<gm32, blk, 0, stream>>>(C, Wm1, nullptr, bm1, T, nullptr, n);
    k_mlp2_softmax<<<gn, blk, 0, stream>>>(T, Wm2, bm2, (float*)d_out, n);
}